// NeighborAggregator_26431228739590
// MI455X (gfx1250) — compile-verified
//
#include <hip/hip_runtime.h>
#include <math.h>

// -----------------------------------------------------------------------------
// NeighborAggregator on MI455X (gfx1250):
//   A_raw[r] = sum over nnz entries with rows==r of adj_values * data[r, c]
//   alpha    = softmax(A_raw)
//   d_out    = [alpha (n) | A_raw (n)]
// Latency-bound random gather + scatter-atomics + tiny softmax. No matmul
// FLOPs -> no WMMA; CDNA5 paths used: NT cache hints, fp32 global atomics,
// async global->LDS staging (ASYNCcnt) for the softmax reduction.
// -----------------------------------------------------------------------------

__global__ __launch_bounds__(256) void na_zero_kernel(float* __restrict__ p, int n) {
    int i = blockIdx.x * blockDim.x + threadIdx.x;
    if (i < n) p[i] = 0.0f;
}

__global__ __launch_bounds__(256) void na_gather_scatter_kernel(
    const float* __restrict__ data,   // [n, n] fp32, 1 GB, random single touches
    const float* __restrict__ vals,   // [nnz] fp32
    const int*   __restrict__ rows,   // [nnz]
    const int*   __restrict__ cols,   // [nnz]
    float*       __restrict__ A_raw,  // [n] fp32, zero-initialized
    int nnz, int n)
{
    int i = blockIdx.x * blockDim.x + threadIdx.x;
    if (i >= nnz) return;
    int   r = rows[i];
    int   c = cols[i];
    float a = vals[i];
    // NT hint: 1 GB matrix, each cacheline touched well under once on average.
    float x = __builtin_nontemporal_load(&data[(size_t)r * (size_t)n + (size_t)c]);
    // 64 KB accumulator region lives in L2; fp32 RMW atomics there are cheap.
    atomicAdd(&A_raw[r], a * x);
}

// Single-workgroup softmax over n == 16384 fp32 values.
// 1024 threads (32 wave32 waves), 16 elements per thread.
__global__ __launch_bounds__(1024) void na_softmax_kernel(
    const float* __restrict__ A_raw,  // [n]
    float*       __restrict__ alpha,  // [n]
    int n)
{
    // Single shared allocation => LDS byte offset 0, which the raw async-copy
    // LDS-address VGPRs below rely on. Layout: [0..n) data, [n..n+32) scratch.
    __shared__ float smem[16384 + 32];
    const int tid = threadIdx.x;
    const int wid = tid >> 5;   // wave id, 0..31
    const int lid = tid & 31;   // lane id in wave32

    // ---- Stage A_raw (64 KB) into LDS via CDNA5 async global->LDS copies ----
    // Each thread issues 4x global_load_async_to_lds_b128 (16 B each):
    //   copies 16 B from global saddr + vaddr to LDS byte address vdst
    // Tracked by ASYNCcnt; completed with s_wait_asynccnt before the barrier.
#pragma unroll
    for (int k = 0; k < 4; ++k) {
        unsigned byteoff = (unsigned)(tid * 16 + k * 16384);
        asm volatile("global_load_async_to_lds_b128 %0, %1, %2"
                     :
                     : "v"(byteoff),   // VDST: LDS byte address (smem base = 0)
                       "v"(byteoff),   // VADDR: 32-bit offset from saddr
                       "s"(A_raw)      // SADDR: 64-bit uniform base
                     : "memory");
    }
    asm volatile("s_wait_asynccnt 0x0" ::: "memory");
    __syncthreads();

    // ---- Pass 1: global max ----
    float lmax = -INFINITY;
#pragma unroll
    for (int k = 0; k < 16; ++k)
        lmax = fmaxf(lmax, smem[tid + k * 1024]);   // lane-stride-1 -> no bank conflicts
#pragma unroll
    for (int off = 16; off >= 1; off >>= 1)
        lmax = fmaxf(lmax, __shfl_xor(lmax, off, 32));
    if (lid == 0) smem[16384 + wid] = lmax;
    __syncthreads();
    float gmax = -INFINITY;
#pragma unroll
    for (int w = 0; w < 32; ++w)
        gmax = fmaxf(gmax, smem[16384 + w]);        // broadcast reads
    __syncthreads();

    // ---- Pass 2: sum of exp(x - gmax) ----
    float lsum = 0.0f;
#pragma unroll
    for (int k = 0; k < 16; ++k)
        lsum += expf(smem[tid + k * 1024] - gmax);
#pragma unroll
    for (int off = 16; off >= 1; off >>= 1)
        lsum += __shfl_xor(lsum, off, 32);
    if (lid == 0) smem[16384 + wid] = lsum;
    __syncthreads();
    float gsum = 0.0f;
#pragma unroll
    for (int w = 0; w < 32; ++w)
        gsum += smem[16384 + w];

    // ---- Pass 3: write alpha ----
    float inv = 1.0f / gsum;
#pragma unroll
    for (int k = 0; k < 16; ++k) {
        int idx = tid + k * 1024;
        alpha[idx] = expf(smem[idx] - gmax) * inv;
    }
}

extern "C" void kernel_launch(void* const* d_in, const int* in_sizes, int n_in,
                              void* d_out, int out_size, void* d_ws, size_t ws_size,
                              hipStream_t stream) {
    const float* data = (const float*)d_in[0];   // [n, n] fp32
    const float* vals = (const float*)d_in[1];   // [nnz] fp32
    const int*   rows = (const int*)d_in[2];     // [nnz] int
    const int*   cols = (const int*)d_in[3];     // [nnz] int

    const int nnz = in_sizes[1];                 // adj_values element count (index-width agnostic)
    const int n   = out_size / 2;                // (alpha, A_raw) concatenated

    float* alpha = (float*)d_out;                // [0, n)
    float* A_raw = (float*)d_out + n;            // [n, 2n)

    // d_out is poisoned and not re-zeroed between replays: zero the accumulator.
    na_zero_kernel<<<(n + 255) / 256, 256, 0, stream>>>(A_raw, n);

    na_gather_scatter_kernel<<<(nnz + 255) / 256, 256, 0, stream>>>(
        data, vals, rows, cols, A_raw, nnz, n);

    na_softmax_kernel<<<1, 1024, 0, stream>>>(A_raw, alpha, n);
}